// Modified3DFisherVectors_78245714199386
// MI455X (gfx1250) — compile-verified
//
#include <hip/hip_runtime.h>
#include <math.h>

typedef float v2f __attribute__((ext_vector_type(2)));
typedef float v8f __attribute__((ext_vector_type(8)));

#define NB       16
#define NPTS     8192
#define G_REAL   125
#define G_PAD    128
#define NCH      20
#define CHUNK    512
#define NWAVES   4
#define PTSW     128      // points per wave
#define PT_TILES 8        // PTSW / 16
#define GT       8        // gaussian tiles (G_PAD/16)
#define LOG2PI   1.8378770664093453f
// -0.5 * log2(e): folded into the B-matrix features so wp = exp2(dot) directly
#define NEGHALF_LOG2E (-0.72134752044448170368f)

#if __has_builtin(__builtin_amdgcn_exp2f)
#define EXP2(x) __builtin_amdgcn_exp2f(x)
#else
#define EXP2(x) exp2f(x)
#endif

// ---------------- global float atomics (CDNA5 native, no-return) -------------
__device__ __forceinline__ void gatomic_max(float* p, float v) {
  asm volatile("global_atomic_max_num_f32 %0, %1, off" :: "v"(p), "v"(v) : "memory");
}
__device__ __forceinline__ void gatomic_min(float* p, float v) {
  asm volatile("global_atomic_min_num_f32 %0, %1, off" :: "v"(p), "v"(v) : "memory");
}
__device__ __forceinline__ void gatomic_add(float* p, float v) {
  asm volatile("global_atomic_add_f32 %0, %1, off" :: "v"(p), "v"(v) : "memory");
}

// ---------------- kernel 0: init accumulators --------------------------------
// channels: 0 pi_max | 1 pi_sum | 2-4 mu_max | 5-7 mu_min | 8-10 mu_sum
//           11-13 sg_max | 14-16 sg_min | 17-19 sg_sum
__global__ void fv_init(float* __restrict__ acc) {
  int i = blockIdx.x * 256 + threadIdx.x;
  if (i >= NB * G_PAD * NCH) return;
  int c = i % NCH;
  bool ismax = (c == 0) || (c >= 2 && c <= 4) || (c >= 11 && c <= 13);
  bool ismin = (c >= 5 && c <= 7) || (c >= 14 && c <= 16);
  acc[i] = ismax ? -INFINITY : (ismin ? INFINITY : 0.0f);
}

// ---------------- kernel 1: main ---------------------------------------------
__launch_bounds__(128)
__global__ void fv_main(const float* __restrict__ points,
                        const float* __restrict__ w,
                        const float* __restrict__ mu,
                        const float* __restrict__ sigma,
                        float* __restrict__ acc) {
  __shared__ float pfeat[CHUNK][8];     // [px,py,pz,px2,py2,pz2,1,0]
  __shared__ float gfeat[G_PAD][8];     // B-matrix features (K=8), pre-scaled by -0.5*log2(e)
  __shared__ float gparam[G_PAD][8];    // [isx,isy,isz, mux*isx,muy*isy,muz*isz, w, 0]
  __shared__ float invsum[CHUNK];       // 1 / sum_g wp

  const int tid  = threadIdx.x;
  const int lane = tid & 31;
  const int wid  = tid >> 5;
  const int hv   = lane >> 4;           // lane half (0/1)
  const int ln   = lane & 15;
  const int b     = blockIdx.y;
  const int chunk = blockIdx.x;
  const int pbase = b * NPTS + chunk * CHUNK;

  // ---- phase 0: stage point & gaussian features in LDS ----
  for (int i = tid; i < CHUNK; i += 128) {
    const float* pp = points + (size_t)(pbase + i) * 3;
    float px = pp[0], py = pp[1], pz = pp[2];
    pfeat[i][0] = px;      pfeat[i][1] = py;      pfeat[i][2] = pz;
    pfeat[i][3] = px * px; pfeat[i][4] = py * py; pfeat[i][5] = pz * pz;
    pfeat[i][6] = 1.0f;    pfeat[i][7] = 0.0f;
  }
  {
    int g = (tid < G_REAL) ? tid : 0;   // pad 125..127 with copies of g0
    float wg = w[g];
    float mx = mu[g*3+0], my = mu[g*3+1], mz = mu[g*3+2];
    float sx = sigma[g*3+0], sy = sigma[g*3+1], sz = sigma[g*3+2];
    float ix = 1.0f/sx, iy = 1.0f/sy, iz = 1.0f/sz;
    float i2x = ix*ix, i2y = iy*iy, i2z = iz*iz;
    float cg = __logf(wg) - (__logf(sx)+__logf(sy)+__logf(sz)) - 1.5f*LOG2PI;
    const float s = NEGHALF_LOG2E;      // wp = exp2(s * mahalanobis_with_const)
    gfeat[tid][0] = s * (-2.0f*mx*i2x);
    gfeat[tid][1] = s * (-2.0f*my*i2y);
    gfeat[tid][2] = s * (-2.0f*mz*i2z);
    gfeat[tid][3] = s * i2x; gfeat[tid][4] = s * i2y; gfeat[tid][5] = s * i2z;
    gfeat[tid][6] = s * ((mx*mx*i2x + my*my*i2y + mz*mz*i2z) - 2.0f*cg);
    gfeat[tid][7] = 0.0f;
    gparam[tid][0] = ix;     gparam[tid][1] = iy;     gparam[tid][2] = iz;
    gparam[tid][3] = mx*ix;  gparam[tid][4] = my*iy;  gparam[tid][5] = mz*iz;
    gparam[tid][6] = wg;     gparam[tid][7] = 0.0f;
  }
  __syncthreads();

  // ---- A fragments for this wave's 8 point tiles (16x4 f32 layout) ----
  // lane holds row M = ln; VGPR pair holds K = 2*hv + {0,1} (frag0), 4+2*hv + {0,1} (frag1)
  const int wbase = wid * PTSW;
  v2f a0[PT_TILES], a1[PT_TILES];
#pragma unroll
  for (int pt = 0; pt < PT_TILES; ++pt) {
    int m = wbase + pt * 16 + ln;
    a0[pt] = *(const v2f*)&pfeat[m][2*hv];
    a1[pt] = *(const v2f*)&pfeat[m][4 + 2*hv];
  }

  // ---- phase 1: posterior denominators via WMMA + exp2 ----
  for (int pt = 0; pt < PT_TILES; ++pt) {
    float sacc[8] = {0,0,0,0,0,0,0,0};
#pragma unroll
    for (int t = 0; t < GT; ++t) {
      int g = t * 16 + ln;
      v2f b0 = *(const v2f*)&gfeat[g][2*hv];
      v2f b1 = *(const v2f*)&gfeat[g][4 + 2*hv];
      v8f d = {};
      d = __builtin_amdgcn_wmma_f32_16x16x4_f32(false, a1[pt], false, b1, (short)0, d, false, false);
      d = __builtin_amdgcn_wmma_f32_16x16x4_f32(false, a0[pt], false, b0, (short)0, d, false, false);
      bool valid = (g < G_REAL);
#pragma unroll
      for (int r = 0; r < 8; ++r) {
        float wp = EXP2(d[r]);                  // = w_g * N(p | mu_g, sigma_g)
        sacc[r] += valid ? wp : 0.0f;
      }
    }
#pragma unroll
    for (int r = 0; r < 8; ++r) {               // reduce over 16 lanes of each half
      float v = sacc[r];
      v += __shfl_xor(v, 1, 32);
      v += __shfl_xor(v, 2, 32);
      v += __shfl_xor(v, 4, 32);
      v += __shfl_xor(v, 8, 32);
      sacc[r] = v;
    }
    if (ln == 0) {                              // lanes 0 and 16 write rows 0-7 / 8-15
      int mb = wbase + pt * 16 + 8 * hv;
#pragma unroll
      for (int r = 0; r < 8; ++r) invsum[mb + r] = 1.0f / sacc[r];
    }
  }
  __syncthreads();

  // ---- phase 2: per-gaussian register accumulators over this wave's points ----
  for (int t = 0; t < GT; ++t) {
    int g = t * 16 + ln;
    float isx = gparam[g][0], isy = gparam[g][1], isz = gparam[g][2];
    float osx = gparam[g][3], osy = gparam[g][4], osz = gparam[g][5];
    float wg  = gparam[g][6];
    v2f b0 = *(const v2f*)&gfeat[g][2*hv];
    v2f b1 = *(const v2f*)&gfeat[g][4 + 2*hv];

    float pi_mx = -INFINITY, pi_sm = 0.0f;
    float mu_mx[3], mu_mn[3], mu_sm[3], sg_mx[3], sg_mn[3], sg_sm[3];
#pragma unroll
    for (int k = 0; k < 3; ++k) {
      mu_mx[k] = -INFINITY; mu_mn[k] = INFINITY; mu_sm[k] = 0.0f;
      sg_mx[k] = -INFINITY; sg_mn[k] = INFINITY; sg_sm[k] = 0.0f;
    }

    for (int pt = 0; pt < PT_TILES; ++pt) {
      v8f d = {};
      d = __builtin_amdgcn_wmma_f32_16x16x4_f32(false, a1[pt], false, b1, (short)0, d, false, false);
      d = __builtin_amdgcn_wmma_f32_16x16x4_f32(false, a0[pt], false, b0, (short)0, d, false, false);
      int mb = wbase + pt * 16 + 8 * hv;
#pragma unroll
      for (int r = 0; r < 8; ++r) {
        int m = mb + r;
        float q  = EXP2(d[r]) * invsum[m];
        float px = pfeat[m][0], py = pfeat[m][1], pz = pfeat[m][2];
        float dx = __builtin_fmaf(px, isx, -osx);
        float dy = __builtin_fmaf(py, isy, -osy);
        float dz = __builtin_fmaf(pz, isz, -osz);
        float cpi = q - wg;
        pi_mx = fmaxf(pi_mx, cpi); pi_sm += cpi;
        float m0 = q * dx, m1 = q * dy, m2 = q * dz;
        mu_mx[0] = fmaxf(mu_mx[0], m0); mu_mn[0] = fminf(mu_mn[0], m0); mu_sm[0] += m0;
        mu_mx[1] = fmaxf(mu_mx[1], m1); mu_mn[1] = fminf(mu_mn[1], m1); mu_sm[1] += m1;
        mu_mx[2] = fmaxf(mu_mx[2], m2); mu_mn[2] = fminf(mu_mn[2], m2); mu_sm[2] += m2;
        float s0 = q * __builtin_fmaf(dx, dx, -1.0f);
        float s1 = q * __builtin_fmaf(dy, dy, -1.0f);
        float s2 = q * __builtin_fmaf(dz, dz, -1.0f);
        sg_mx[0] = fmaxf(sg_mx[0], s0); sg_mn[0] = fminf(sg_mn[0], s0); sg_sm[0] += s0;
        sg_mx[1] = fmaxf(sg_mx[1], s1); sg_mn[1] = fminf(sg_mn[1], s1); sg_sm[1] += s1;
        sg_mx[2] = fmaxf(sg_mx[2], s2); sg_mn[2] = fminf(sg_mn[2], s2); sg_sm[2] += s2;
      }
    }

    // combine the two lane-halves (same gaussian, disjoint point rows)
    pi_mx = fmaxf(pi_mx, __shfl_xor(pi_mx, 16, 32));
    pi_sm +=              __shfl_xor(pi_sm, 16, 32);
#pragma unroll
    for (int k = 0; k < 3; ++k) {
      mu_mx[k] = fmaxf(mu_mx[k], __shfl_xor(mu_mx[k], 16, 32));
      mu_mn[k] = fminf(mu_mn[k], __shfl_xor(mu_mn[k], 16, 32));
      mu_sm[k] +=                __shfl_xor(mu_sm[k], 16, 32);
      sg_mx[k] = fmaxf(sg_mx[k], __shfl_xor(sg_mx[k], 16, 32));
      sg_mn[k] = fminf(sg_mn[k], __shfl_xor(sg_mn[k], 16, 32));
      sg_sm[k] +=                __shfl_xor(sg_sm[k], 16, 32);
    }

    if (hv == 0 && g < G_REAL) {
      float* row = acc + ((size_t)b * G_PAD + g) * NCH;
      gatomic_max(row + 0, pi_mx);
      gatomic_add(row + 1, pi_sm);
#pragma unroll
      for (int k = 0; k < 3; ++k) {
        gatomic_max(row + 2  + k, mu_mx[k]);
        gatomic_min(row + 5  + k, mu_mn[k]);
        gatomic_add(row + 8  + k, mu_sm[k]);
        gatomic_max(row + 11 + k, sg_mx[k]);
        gatomic_min(row + 14 + k, sg_mn[k]);
        gatomic_add(row + 17 + k, sg_sm[k]);
      }
    }
  }
}

// ---------------- kernel 2: scale + power norm + L2 norm + transpose ---------
__launch_bounds__(128)
__global__ void fv_norm(const float* __restrict__ acc,
                        const float* __restrict__ w,
                        float* __restrict__ out) {
  __shared__ float ssum[NCH];
  int b = blockIdx.x;
  int tid = threadIdx.x;
  if (tid < NCH) ssum[tid] = 0.0f;
  __syncthreads();

  float vals[NCH];
  if (tid < G_REAL) {
    float wg = w[tid];
    float rn  = 1.0f / (float)NPTS;
    float spi = rn * rsqrtf(wg);          // also d_mu scale
    float ssg = rn * rsqrtf(2.0f * wg);
    const float* row = acc + ((size_t)b * G_PAD + tid) * NCH;
#pragma unroll
    for (int c = 0; c < NCH; ++c) {
      float scale = (c < 2) ? spi : ((c < 11) ? spi : ssg);
      float x = row[c] * scale;
      float v = copysignf(sqrtf(fabsf(x)), x);   // sign(x)*|x|^0.5
      vals[c] = v;
      atomicAdd(&ssum[c], v * v);
    }
  }
  __syncthreads();
  if (tid < G_REAL) {
#pragma unroll
    for (int c = 0; c < NCH; ++c) {
      float rnorm = rsqrtf(fmaxf(ssum[c], 1e-12f));
      out[((size_t)b * NCH + c) * G_REAL + tid] = vals[c] * rnorm;
    }
  }
}

// ---------------- launch -----------------------------------------------------
extern "C" void kernel_launch(void* const* d_in, const int* in_sizes, int n_in,
                              void* d_out, int out_size, void* d_ws, size_t ws_size,
                              hipStream_t stream) {
  const float* points = (const float*)d_in[0];   // [16,8192,3]
  const float* w      = (const float*)d_in[1];   // [125]
  const float* mu     = (const float*)d_in[2];   // [125,3]
  const float* sigma  = (const float*)d_in[3];   // [125,3]
  float* out = (float*)d_out;                    // [16,20,125]
  float* acc = (float*)d_ws;                     // [16][128][20] f32 = 160 KB

  int n_init = NB * G_PAD * NCH;
  fv_init<<<(n_init + 255) / 256, 256, 0, stream>>>(acc);
  dim3 grid(NPTS / CHUNK, NB);                   // 16 x 16 workgroups
  fv_main<<<grid, 128, 0, stream>>>(points, w, mu, sigma, acc);
  fv_norm<<<NB, 128, 0, stream>>>(acc, w, out);
}